// RGCN_PyG_19585050870242
// MI455X (gfx1250) — compile-verified
//
#include <hip/hip_runtime.h>
#include <hip/hip_bf16.h>

#define NNODES 50000
#define NEDGES 1600000
#define DF     64
#define NREL   8

typedef __attribute__((ext_vector_type(2))) float v2f;
typedef __attribute__((ext_vector_type(8))) float v8f;

// ---------------------------------------------------------------------------
// Node pre-transform with fp32 WMMA:
//   h[r][n][:]   = x[n] @ W[r]        for r = 0..7
//   out_root[n]  = x[n] @ root + bias (initializes the aggregation buffer)
// One block (256 thr = 8 waves) per 16-node tile. 9 matrices * 4 col-tiles
// = 36 16x16 output tiles split across the 8 waves. K=64 -> 16 chained
// v_wmma_f32_16x16x4_f32 per tile.
// ---------------------------------------------------------------------------
__global__ __launch_bounds__(256) void rgcn_transform(
    const float* __restrict__ x,       // [N, D]
    const float* __restrict__ W,       // [R, D, D]
    const float* __restrict__ root,    // [D, D]
    const float* __restrict__ bias,    // [D]
    float* __restrict__ h,             // [R, N, D] (workspace)
    float* __restrict__ out_root,      // [N, D]
    int relu_input)
{
    __shared__ float xs[16][DF + 1];   // +1 pad: stride 65 kills bank conflicts

    const int tid      = threadIdx.x;
    const int nodebase = blockIdx.x * 16;

    // Cooperative stage of the 16x64 fp32 x-tile (ReLU folded in for layer 2).
    for (int idx = tid; idx < 16 * DF; idx += 256) {
        const int rr = idx >> 6, cc = idx & 63;
        float v = x[(size_t)(nodebase + rr) * DF + cc];
        if (relu_input) v = fmaxf(v, 0.0f);
        xs[rr][cc] = v;
    }
    __syncthreads();

    const int lane = tid & 31;
    const int wave = tid >> 5;
    const int half = lane >> 4;        // K-half select per ISA A/B layout
    const int m    = lane & 15;        // row (A) / column (B,C,D) index

    // A fragments for all 16 K-steps (ISA 7.12.2, 32-bit A 16x4):
    //   lanes 0-15 : v0=K0, v1=K1 ; lanes 16-31 : v0=K2, v1=K3
    v2f a[16];
#pragma unroll
    for (int s = 0; s < 16; ++s) {
        a[s].x = xs[m][4 * s + 2 * half];
        a[s].y = xs[m][4 * s + 2 * half + 1];
    }

    // 36 output tiles, wave-uniform loop -> EXEC all-ones at every WMMA.
    for (int t = wave; t < 36; t += 8) {
        const int r       = t >> 2;            // 0..7 = relation, 8 = root
        const int colbase = (t & 3) * 16;
        const float* Wp   = (r < NREL) ? (W + (size_t)r * DF * DF) : root;

        v8f c = {};
#pragma unroll
        for (int s = 0; s < 16; ++s) {
            // B 4x16 fp32 (mirror of A): lanes 0-15 rows K0/K1, 16-31 K2/K3
            v2f b;
            b.x = Wp[(size_t)(4 * s + 2 * half) * DF + colbase + m];
            b.y = Wp[(size_t)(4 * s + 2 * half + 1) * DF + colbase + m];
            c = __builtin_amdgcn_wmma_f32_16x16x4_f32(
                    /*neg_a=*/false, a[s], /*neg_b=*/false, b,
                    /*c_mod=*/(short)0, c, /*reuse_a=*/false, /*reuse_b=*/false);
        }

        // C/D layout: VGPR i -> lanes 0-15: M=i, lanes 16-31: M=i+8, N=m.
        if (r < NREL) {
            float* outp = h + ((size_t)r * NNODES + nodebase) * DF + colbase + m;
#pragma unroll
            for (int i = 0; i < 8; ++i)
                outp[(size_t)(i + 8 * half) * DF] = c[i];
        } else {
            const float bv = bias[colbase + m];
            float* outp = out_root + (size_t)nodebase * DF + colbase + m;
#pragma unroll
            for (int i = 0; i < 8; ++i)
                outp[(size_t)(i + 8 * half) * DF] = c[i] + bv;
        }
    }
}

// ---------------------------------------------------------------------------
// Edge scatter-add: out[dst] += h[etype, src]. 16 lanes per edge, float4
// gather + 4 global f32 atomics per lane. Memory/atomic bound.
// ---------------------------------------------------------------------------
__global__ __launch_bounds__(256) void rgcn_aggregate(
    const int* __restrict__ src,
    const int* __restrict__ dst,
    const int* __restrict__ etype,
    const float* __restrict__ h,       // [R, N, D]
    float* __restrict__ out,           // [N, D]
    int n_edges)
{
    const int gtid   = blockIdx.x * 256 + threadIdx.x;
    const int edge   = gtid >> 4;
    const int lane16 = gtid & 15;
    if (edge >= n_edges) return;

    const int r = etype[edge];
    const int s = src[edge];
    const int d = dst[edge];

    const float4 v = ((const float4*)(h + ((size_t)r * NNODES + s) * DF))[lane16];
    float* op = out + (size_t)d * DF + lane16 * 4;
    atomicAdd(op + 0, v.x);
    atomicAdd(op + 1, v.y);
    atomicAdd(op + 2, v.z);
    atomicAdd(op + 3, v.w);
}

extern "C" void kernel_launch(void* const* d_in, const int* in_sizes, int n_in,
                              void* d_out, int out_size, void* d_ws, size_t ws_size,
                              hipStream_t stream) {
    const int*   adj      = (const int*)  d_in[0];   // [2, E]
    const float* features = (const float*)d_in[1];   // [N, D]
    const int*   etype    = (const int*)  d_in[2];   // [E]
    const float* W1       = (const float*)d_in[3];   // [R, D, D]
    const float* root1    = (const float*)d_in[4];   // [D, D]
    const float* b1       = (const float*)d_in[5];   // [D]
    const float* W2       = (const float*)d_in[6];
    const float* root2    = (const float*)d_in[7];
    const float* b2       = (const float*)d_in[8];
    float*       out      = (float*)d_out;           // [N, D]

    const int* srcp = adj;
    const int* dstp = adj + NEDGES;

    // Workspace: h (R*N*D fp32 = 102.4 MB) then x1 (N*D fp32 = 12.8 MB).
    float* h  = (float*)d_ws;
    float* x1 = h + (size_t)NREL * NNODES * DF;

    const dim3 tblk(256);
    const dim3 tgrid(NNODES / 16);                       // 3125 tiles
    const dim3 agrid((NEDGES * 16 + 255) / 256);         // 16 lanes/edge

    // Layer 1: root-term init of x1, then edge scatter into x1.
    rgcn_transform<<<tgrid, tblk, 0, stream>>>(features, W1, root1, b1, h, x1, 0);
    rgcn_aggregate<<<agrid, tblk, 0, stream>>>(srcp, dstp, etype, h, x1, NEDGES);

    // Layer 2: ReLU folded into the x1 read; root term writes d_out directly.
    rgcn_transform<<<tgrid, tblk, 0, stream>>>(x1, W2, root2, b2, h, out, 1);
    rgcn_aggregate<<<agrid, tblk, 0, stream>>>(srcp, dstp, etype, h, out, NEDGES);
}